// ResidualVectorQuantizer_32641751450046
// MI455X (gfx1250) — compile-verified
//
#include <hip/hip_runtime.h>

// CDNA5 wave32 fp32 WMMA fragments
typedef __attribute__((ext_vector_type(2))) float v2f;
typedef __attribute__((ext_vector_type(8))) float v8f;

#define CB_K   1024
#define EDIM   128
#define NCB    8
#define PITCH  132   // 128 + 4 floats pad -> conflict-free column reads (64 banks)

// ---------------- codebook squared-norm precompute (into d_ws) ----------------
__global__ void rvq_cbnorm_kernel(const float* __restrict__ cbs,
                                  float* __restrict__ cbnorm) {
  int row = blockIdx.x * blockDim.x + threadIdx.x;     // 0 .. 8191
  if (row < NCB * CB_K) {
    const float* p = cbs + (size_t)row * EDIM;
    float s = 0.f;
#pragma unroll 8
    for (int e = 0; e < EDIM; ++e) s = fmaf(p[e], p[e], s);
    cbnorm[row] = s;
  }
}

// ---------------- main RVQ kernel: one wave per 16-row tile ----------------
__global__ void __launch_bounds__(32)
rvq_main_kernel(const float* __restrict__ emb,     // [16,128,64,64] f32
                const float* __restrict__ cbs,     // [8,1024,128]   f32
                const float* __restrict__ cbnorm,  // [8*1024]       f32
                float* __restrict__ out) {         // [16,128,64,64] f32
  __shared__ float sres[16 * PITCH];
  __shared__ float squant[16 * PITCH];
  __shared__ int   sidx[16];

  const int lane  = threadIdx.x;    // 0..31 (wave32)
  const int mlo   = lane & 15;
  const int khalf = lane >> 4;      // 0: K pair {0,1}, 1: K pair {2,3}

  // tile covers rows n0..n0+15 of x, n = (b*64 + h)*64 + w
  const int n0 = blockIdx.x * 16;
  const int b  = n0 >> 12;
  const int h  = (n0 >> 6) & 63;
  const int w0 = n0 & 63;           // multiple of 16 -> same (b,h) for whole tile
  const size_t gbase = (size_t)b * (EDIM * 64 * 64) + (size_t)h * 64 + (size_t)w0;

  // load residual tile (lane = w -> coalesced), zero quant accumulator
  for (int t = lane; t < 16 * EDIM; t += 32) {
    int m = t & 15, e = t >> 4;
    sres[m * PITCH + e]   = emb[gbase + (size_t)e * 4096 + m];
    squant[m * PITCH + e] = 0.f;
  }
  __syncthreads();

  for (int i = 0; i < NCB; ++i) {
    const float* cbi  = cbs + (size_t)i * CB_K * EDIM;
    const float* cbni = cbnorm + i * CB_K;

    float bestv[8];
    int   besti[8];
#pragma unroll
    for (int r = 0; r < 8; ++r) { bestv[r] = 3.4e38f; besti[r] = 0; }

    for (int j0 = 0; j0 < CB_K; j0 += 16) {
      v8f acc = {0.f, 0.f, 0.f, 0.f, 0.f, 0.f, 0.f, 0.f};
      // A frag: sres[mlo][k0 + 2*khalf + {0,1}]   (ISA 16x4 f32 A layout)
      // B frag: cb[j0+mlo][k0 + 2*khalf + {0,1}]  (4x16 f32 B layout, N = mlo)
      const float* arow = sres + mlo * PITCH + 2 * khalf;
      const float* brow = cbi + (size_t)(j0 + mlo) * EDIM + 2 * khalf;
#pragma unroll
      for (int k0 = 0; k0 < EDIM; k0 += 4) {
        v2f a, bf;
        a.x  = arow[k0];  a.y  = arow[k0 + 1];
        bf.x = brow[k0];  bf.y = brow[k0 + 1];
        acc = __builtin_amdgcn_wmma_f32_16x16x4_f32(
                /*neg_a=*/false, a, /*neg_b=*/false, bf,
                /*c_mod=*/(short)0, acc, /*reuse_a=*/false, /*reuse_b=*/false);
      }
      const float cbn  = cbni[j0 + mlo];
      const int   jidx = j0 + mlo;
#pragma unroll
      for (int r = 0; r < 8; ++r) {
        float score = fmaf(-2.f, acc[r], cbn);   // ||c||^2 - 2 r.c
        if (score < bestv[r]) { bestv[r] = score; besti[r] = jidx; }
      }
    }

    // argmin across the 16 lanes of each half-wave (row M = r + 8*khalf)
#pragma unroll
    for (int r = 0; r < 8; ++r) {
      float v  = bestv[r];
      int   id = besti[r];
#pragma unroll
      for (int mask = 1; mask < 16; mask <<= 1) {
        float ov = __shfl_xor(v, mask, 16);
        int   oi = __shfl_xor(id, mask, 16);
        if (ov < v || (ov == v && oi < id)) { v = ov; id = oi; }
      }
      if (mlo == 0) sidx[r + 8 * khalf] = id;
    }
    __syncthreads();

    // gather decode (coalesced over e) + residual/quant update
    for (int t = lane; t < 16 * EDIM; t += 32) {
      int m = t >> 7, e = t & 127;
      float dv = cbi[(size_t)sidx[m] * EDIM + e];
      sres[m * PITCH + e]   -= dv;
      squant[m * PITCH + e] += dv;
    }
    __syncthreads();
  }

  // write quantized tile back with transpose (lane = w -> coalesced)
  for (int t = lane; t < 16 * EDIM; t += 32) {
    int m = t & 15, e = t >> 4;
    out[gbase + (size_t)e * 4096 + m] = squant[m * PITCH + e];
  }
}

extern "C" void kernel_launch(void* const* d_in, const int* in_sizes, int n_in,
                              void* d_out, int out_size, void* d_ws, size_t ws_size,
                              hipStream_t stream) {
  (void)n_in; (void)out_size; (void)ws_size;
  const float* emb = (const float*)d_in[0];   // (16,128,64,64) f32
  const float* cbs = (const float*)d_in[1];   // (8,1024,128)   f32
  float* out    = (float*)d_out;
  float* cbnorm = (float*)d_ws;               // 8192 floats of scratch

  rvq_cbnorm_kernel<<<dim3((NCB * CB_K + 255) / 256), dim3(256), 0, stream>>>(cbs, cbnorm);

  const int nrows   = in_sizes[0] / EDIM;     // 65536
  const int nblocks = nrows / 16;             // 4096 single-wave workgroups
  rvq_main_kernel<<<dim3(nblocks), dim3(32), 0, stream>>>(emb, cbs, cbnorm, out);
}